// SAGEMalwareClassifier_84129819394427
// MI455X (gfx1250) — compile-verified
//
#include <hip/hip_runtime.h>

typedef float v2f __attribute__((ext_vector_type(2)));
typedef float v8f __attribute__((ext_vector_type(8)));

#define N_NODES    100000
#define N_EDGES    3200000
#define IN_CH      128
#define HIDDEN     64
#define NUM_GRAPHS 128
#define BN_EPS     1e-5f

// ---------------- degree count + reciprocal ----------------
__global__ void count_deg_kernel(const int* __restrict__ dst, float* __restrict__ cnt, int nE) {
    int gid = blockIdx.x * blockDim.x + threadIdx.x;
    if (gid < nE) atomicAdd(&cnt[dst[gid]], 1.0f);
}

__global__ void recip_kernel(float* __restrict__ c, int n) {
    int gid = blockIdx.x * blockDim.x + threadIdx.x;
    if (gid < n) c[gid] = 1.0f / fmaxf(c[gid], 1.0f);
}

// ---------------- edge scatter-add (mean numerator) ----------------
// One thread per (edge, channel); exact grid: nE*C / 256 blocks.
template<int C, int LOGC>
__global__ void scatter_kernel(const float* __restrict__ xin,
                               const int* __restrict__ src,
                               const int* __restrict__ dst,
                               float* __restrict__ agg) {
    int gid = blockIdx.x * blockDim.x + threadIdx.x;
    int e = gid >> LOGC;
    int c = gid & (C - 1);
    int s = src[e];
    int d = dst[e];
    atomicAdd(&agg[d * C + c], xin[s * C + c]);
}

// ---------------- fused SAGE linear via fp32 WMMA + BN stats ----------------
// Block = 128 threads = 4 waves; block owns 16 nodes; wave w owns cols [16w,16w+16).
// h_pre = (aggsum * invcnt) @ Wl + bl + hin @ Wr ; also accumulates column sum/sumsq.
template<int C>
__global__ void __launch_bounds__(128)
sage_gemm_kernel(const float* __restrict__ aggsum,
                 const float* __restrict__ invcnt,
                 const float* __restrict__ hin,
                 const float* __restrict__ Wl,
                 const float* __restrict__ Wr,
                 const float* __restrict__ bl,
                 float* __restrict__ hpre,
                 float* __restrict__ stats /* [0,64)=sum, [64,128)=sumsq */) {
    const int lane     = threadIdx.x & 31;
    const int wave     = threadIdx.x >> 5;
    const int nodeBase = blockIdx.x * 16;
    const int m        = lane & 15;   // row within A tile / col within B,D tile
    const int kp       = lane >> 4;   // K-pair select (0: K{0,1}, 1: K{2,3})
    const int colBase  = wave * 16;
    const int col      = colBase + m;

    const int   rowA = nodeBase + m;
    const float inv  = invcnt[rowA];

    const float* __restrict__ arow = aggsum + rowA * C + 2 * kp;
    const float* __restrict__ xrow = hin    + rowA * C + 2 * kp;

    v8f acc = {};
    // GEMM 1: mean-aggregated neighbors @ Wl
    for (int k = 0; k < C; k += 4) {
        v2f a;
        a.x = arow[k]     * inv;
        a.y = arow[k + 1] * inv;
        v2f b;
        b.x = Wl[(k + 2 * kp)     * HIDDEN + col];
        b.y = Wl[(k + 2 * kp + 1) * HIDDEN + col];
        acc = __builtin_amdgcn_wmma_f32_16x16x4_f32(false, a, false, b,
                                                    (short)0, acc, false, false);
    }
    // GEMM 2: self features @ Wr
    for (int k = 0; k < C; k += 4) {
        v2f a;
        a.x = xrow[k];
        a.y = xrow[k + 1];
        v2f b;
        b.x = Wr[(k + 2 * kp)     * HIDDEN + col];
        b.y = Wr[(k + 2 * kp + 1) * HIDDEN + col];
        acc = __builtin_amdgcn_wmma_f32_16x16x4_f32(false, a, false, b,
                                                    (short)0, acc, false, false);
    }

    const float bias = bl[col];
    float s = 0.0f, sq = 0.0f;
#pragma unroll
    for (int i = 0; i < 8; ++i) {
        int   row = nodeBase + i + 8 * kp;       // D layout: vgpr i -> M=i (+8 for upper lanes)
        float v   = acc[i] + bias;
        hpre[row * HIDDEN + col] = v;
        s  += v;
        sq += v * v;
    }
    // fold the two lane halves (same column) then one atomic per column per block
    s  += __shfl_xor(s, 16, 32);
    sq += __shfl_xor(sq, 16, 32);
    if (kp == 0) {
        atomicAdd(&stats[col], s);
        atomicAdd(&stats[HIDDEN + col], sq);
    }
}

// ---------------- BN finalize: scale/shift from column sums ----------------
__global__ void bn_finalize_kernel(const float* __restrict__ stats,
                                   const float* __restrict__ gamma,
                                   const float* __restrict__ beta,
                                   float* __restrict__ bnp /* [0,64)=scale, [64,128)=shift */) {
    int c = threadIdx.x;  // 64 threads
    const float invN = 1.0f / (float)N_NODES;
    float mu  = stats[c] * invN;
    float var = stats[HIDDEN + c] * invN - mu * mu;
    float rs  = rsqrtf(var + BN_EPS);
    float sc  = gamma[c] * rs;
    bnp[c]          = sc;
    bnp[HIDDEN + c] = beta[c] - mu * sc;
}

// ---------------- apply BN + ReLU ----------------
__global__ void bn_relu_kernel(const float* __restrict__ hpre,
                               const float* __restrict__ bnp,
                               float* __restrict__ hout) {
    int gid = blockIdx.x * blockDim.x + threadIdx.x;   // exact: N_NODES*64/256 blocks
    int c = gid & (HIDDEN - 1);
    float v = hpre[gid] * bnp[c] + bnp[HIDDEN + c];
    hout[gid] = fmaxf(v, 0.0f);
}

// ---------------- global mean pool (atomics) ----------------
__global__ void pool_kernel(const float* __restrict__ h,
                            const int* __restrict__ batch,
                            float* __restrict__ gsum,
                            float* __restrict__ gcnt) {
    int gid = blockIdx.x * blockDim.x + threadIdx.x;   // N_NODES*64 threads
    int n = gid >> 6;
    int c = gid & 63;
    int b = batch[n];
    atomicAdd(&gsum[b * HIDDEN + c], h[gid]);
    if (c == 0) atomicAdd(&gcnt[b], 1.0f);
}

// ---------------- classifier MLP (128 graphs, 1 block) ----------------
__global__ void __launch_bounds__(128)
classifier_kernel(const float* __restrict__ gsum,
                  const float* __restrict__ gcnt,
                  const float* __restrict__ W1, const float* __restrict__ b1,
                  const float* __restrict__ W2, const float* __restrict__ b2,
                  float* __restrict__ out) {
    int g = threadIdx.x;  // 128 threads
    float ic = 1.0f / fmaxf(gcnt[g], 1.0f);
    float feat[HIDDEN];
#pragma unroll
    for (int c = 0; c < HIDDEN; ++c) feat[c] = gsum[g * HIDDEN + c] * ic;
    float o0 = b2[0], o1 = b2[1];
#pragma unroll 4
    for (int j = 0; j < HIDDEN / 2; ++j) {
        float hj = b1[j];
#pragma unroll
        for (int c = 0; c < HIDDEN; ++c) hj += feat[c] * W1[c * (HIDDEN / 2) + j];
        hj = fmaxf(hj, 0.0f);
        o0 += hj * W2[j * 2 + 0];
        o1 += hj * W2[j * 2 + 1];
    }
    out[g * 2 + 0] = o0;
    out[g * 2 + 1] = o1;
}

// ---------------- host orchestration ----------------
static inline size_t align_up(size_t v, size_t a) { return (v + a - 1) & ~(a - 1); }

extern "C" void kernel_launch(void* const* d_in, const int* in_sizes, int n_in,
                              void* d_out, int out_size, void* d_ws, size_t ws_size,
                              hipStream_t stream) {
    const float* x     = (const float*)d_in[0];
    const int*   ei    = (const int*)d_in[1];
    const int*   srcp  = ei;
    const int*   dstp  = ei + N_EDGES;
    const int*   batch = (const int*)d_in[2];

    const float* Wl[3]    = {(const float*)d_in[3],  (const float*)d_in[8],  (const float*)d_in[13]};
    const float* blv[3]   = {(const float*)d_in[4],  (const float*)d_in[9],  (const float*)d_in[14]};
    const float* Wr[3]    = {(const float*)d_in[5],  (const float*)d_in[10], (const float*)d_in[15]};
    const float* gam[3]   = {(const float*)d_in[6],  (const float*)d_in[11], (const float*)d_in[16]};
    const float* bet[3]   = {(const float*)d_in[7],  (const float*)d_in[12], (const float*)d_in[17]};
    const float* W1 = (const float*)d_in[18];
    const float* b1 = (const float*)d_in[19];
    const float* W2 = (const float*)d_in[20];
    const float* b2 = (const float*)d_in[21];

    // workspace carve-out
    char*  ws  = (char*)d_ws;
    size_t off = 0;
    float* aggsum = (float*)(ws + off); off = align_up(off + (size_t)N_NODES * IN_CH  * 4, 256);
    float* hbuf   = (float*)(ws + off); off = align_up(off + (size_t)N_NODES * HIDDEN * 4, 256);
    float* hpre   = (float*)(ws + off); off = align_up(off + (size_t)N_NODES * HIDDEN * 4, 256);
    float* invcnt = (float*)(ws + off); off = align_up(off + (size_t)N_NODES * 4, 256);
    float* stats  = (float*)(ws + off); off = align_up(off + 128 * 4, 256);
    float* bnp    = (float*)(ws + off); off = align_up(off + 128 * 4, 256);
    float* gsum   = (float*)(ws + off); off = align_up(off + (size_t)NUM_GRAPHS * HIDDEN * 4, 256);
    float* gcnt   = (float*)(ws + off); off = align_up(off + NUM_GRAPHS * 4, 256);
    (void)ws_size; (void)in_sizes; (void)n_in; (void)out_size;

    // degree -> 1/max(deg,1), computed once (dst is layer-invariant)
    hipMemsetAsync(invcnt, 0, (size_t)N_NODES * 4, stream);
    count_deg_kernel<<<(N_EDGES + 255) / 256, 256, 0, stream>>>(dstp, invcnt, N_EDGES);
    recip_kernel<<<(N_NODES + 255) / 256, 256, 0, stream>>>(invcnt, N_NODES);

    const float* hin = x;
    for (int layer = 0; layer < 3; ++layer) {
        const int C = (layer == 0) ? IN_CH : HIDDEN;
        hipMemsetAsync(aggsum, 0, (size_t)N_NODES * C * 4, stream);
        hipMemsetAsync(stats,  0, 128 * 4, stream);

        if (C == IN_CH) {
            scatter_kernel<IN_CH, 7><<<(size_t)N_EDGES * IN_CH / 256, 256, 0, stream>>>(
                hin, srcp, dstp, aggsum);
            sage_gemm_kernel<IN_CH><<<N_NODES / 16, 128, 0, stream>>>(
                aggsum, invcnt, hin, Wl[layer], Wr[layer], blv[layer], hpre, stats);
        } else {
            scatter_kernel<HIDDEN, 6><<<(size_t)N_EDGES * HIDDEN / 256, 256, 0, stream>>>(
                hin, srcp, dstp, aggsum);
            sage_gemm_kernel<HIDDEN><<<N_NODES / 16, 128, 0, stream>>>(
                aggsum, invcnt, hin, Wl[layer], Wr[layer], blv[layer], hpre, stats);
        }
        bn_finalize_kernel<<<1, HIDDEN, 0, stream>>>(stats, gam[layer], bet[layer], bnp);
        bn_relu_kernel<<<(size_t)N_NODES * HIDDEN / 256, 256, 0, stream>>>(hpre, bnp, hbuf);
        hin = hbuf;  // hbuf free to overwrite next iteration (read fully before write)
    }

    // global mean pool + classifier
    hipMemsetAsync(gsum, 0, (size_t)NUM_GRAPHS * HIDDEN * 4, stream);
    hipMemsetAsync(gcnt, 0, (size_t)NUM_GRAPHS * 4, stream);
    pool_kernel<<<(size_t)N_NODES * HIDDEN / 256, 256, 0, stream>>>(hbuf, batch, gsum, gcnt);
    classifier_kernel<<<1, NUM_GRAPHS, 0, stream>>>(gsum, gcnt, W1, b1, W2, b2, (float*)d_out);
}